// MultiCameraTransform_4964982194693
// MI455X (gfx1250) — compile-verified
//
#include <hip/hip_runtime.h>

typedef __attribute__((ext_vector_type(2))) float v2f;
typedef __attribute__((ext_vector_type(8))) float v8f;

#define SE3_PAIRS 2   // 2 pairs = 4 tiles = 64 columns per wave per iteration

// Swap the two 16-lane halves of a wave32 register (identity nibble selects).
__device__ __forceinline__ float swap_halves(float f) {
    unsigned u = __builtin_bit_cast(unsigned, f);
    unsigned r = __builtin_amdgcn_permlanex16(u, u, 0x76543210u, 0xfedcba98u, false, false);
    return __builtin_bit_cast(float, r);
}

__global__ __launch_bounds__(256) void se3_wmma_kernel(
    const float* __restrict__ x,      // [4, N]
    const float* __restrict__ wv,     // [16, 3]
    const float* __restrict__ vv,     // [16, 3]
    const float* __restrict__ thv,    // [16, 1]
    const int*   __restrict__ cip,    // scalar camera index
    float* __restrict__ out,          // [4, N]
    int N)
{
    // ---- Build E = [[R, t],[0,0,0,1]] once per block (thread 0), share via LDS ----
    __shared__ float sE[16];
    if (threadIdx.x == 0) {
        const int ci = *cip;
        const float w0 = wv[3*ci+0], w1 = wv[3*ci+1], w2 = wv[3*ci+2];
        const float v0 = vv[3*ci+0], v1 = vv[3*ci+1], v2 = vv[3*ci+2];
        const float th = thv[ci];
        const float s = __sinf(th), c = __cosf(th);
        const float omc = 1.0f - c;
        const float tms = th - s;
        const float ww  = w0*w0 + w1*w1 + w2*w2;
        const float K01 = -w2, K02 =  w1;
        const float K10 =  w2, K12 = -w0;
        const float K20 = -w1, K21 =  w0;
        const float K200 = w0*w0 - ww, K201 = w0*w1,      K202 = w0*w2;
        const float K210 = w1*w0,      K211 = w1*w1 - ww, K212 = w1*w2;
        const float K220 = w2*w0,      K221 = w2*w1,      K222 = w2*w2 - ww;
        // R = I + s K + (1-c) K^2
        sE[0*4+0] = 1.0f         + omc*K200;
        sE[0*4+1] =        s*K01 + omc*K201;
        sE[0*4+2] =        s*K02 + omc*K202;
        sE[1*4+0] =        s*K10 + omc*K210;
        sE[1*4+1] = 1.0f         + omc*K211;
        sE[1*4+2] =        s*K12 + omc*K212;
        sE[2*4+0] =        s*K20 + omc*K220;
        sE[2*4+1] =        s*K21 + omc*K221;
        sE[2*4+2] = 1.0f         + omc*K222;
        // t = (I*th + (1-c)K + (th-s)K^2) @ v
        const float V00 = th          + tms*K200, V01 = omc*K01 + tms*K201, V02 = omc*K02 + tms*K202;
        const float V10 = omc*K10 + tms*K210, V11 = th          + tms*K211, V12 = omc*K12 + tms*K212;
        const float V20 = omc*K20 + tms*K220, V21 = omc*K21 + tms*K221, V22 = th          + tms*K222;
        sE[0*4+3] = V00*v0 + V01*v1 + V02*v2;
        sE[1*4+3] = V10*v0 + V11*v1 + V12*v2;
        sE[2*4+3] = V20*v0 + V21*v1 + V22*v2;
        sE[3*4+0] = 0.0f; sE[3*4+1] = 0.0f; sE[3*4+2] = 0.0f; sE[3*4+3] = 1.0f;
    }
    __syncthreads();

    const int lane = (int)(threadIdx.x & 31);
    const int m    = lane & 15;          // row/column within lane-half
    const int kk   = (lane >> 4) << 1;   // K pair: 0 (lanes 0-15), 2 (lanes 16-31)
    const bool lo  = (lane < 16);

    // A matrix (16x4 f32): rows 0-3 = E, rows 4-15 = 0
    v2f a;
    a.x = (m < 4) ? sE[m*4 + kk + 0] : 0.0f;
    a.y = (m < 4) ? sE[m*4 + kk + 1] : 0.0f;

    const int waveId     = (int)((blockIdx.x * blockDim.x + threadIdx.x) >> 5);
    const int numWaves   = (int)((gridDim.x * blockDim.x) >> 5);
    const int nTiles     = (N + 15) >> 4;
    const int tilesPerIt = 2 * SE3_PAIRS;

    // Signed 32-bit element offsets (max 4*N < 2^25): signed no-wrap lets the
    // compiler emit SADDR-form VMEM (uniform base + VGPR_I32 offset, scale_offset).
    const int rowBase = kk * N;   // row 0 or row 2 base for this lane-half

    for (int tb = waveId * tilesPerIt; tb < nTiles; tb += numWaves * tilesPerIt) {
        // Prefetch next grid-stride iteration's B operands.
        {
            const int tn = tb + numWaves * tilesPerIt;
            if (tn < nTiles) {
                const int cc = (tn << 4) + m;
                __builtin_prefetch(x + rowBase + cc, 0, 1);
                __builtin_prefetch(x + rowBase + N + cc, 0, 1);
            }
        }

        #pragma unroll
        for (int p = 0; p < SE3_PAIRS; ++p) {
            const int t0 = tb + 2 * p;

            // ---- Load B operands for both tiles of the pair (NT streaming) ----
            v2f b0, b1;
            {
                const int col0 = (t0 << 4) + m;
                const bool in0 = col0 < N;
                const int cc0 = in0 ? col0 : 0;
                float bx0 = __builtin_nontemporal_load(x + rowBase + cc0);
                float by0 = __builtin_nontemporal_load(x + rowBase + N + cc0);
                b0.x = in0 ? bx0 : 0.0f;
                b0.y = in0 ? by0 : 0.0f;

                const int col1 = col0 + 16;
                const bool in1 = col1 < N;
                const int cc1 = in1 ? col1 : 0;
                float bx1 = __builtin_nontemporal_load(x + rowBase + cc1);
                float by1 = __builtin_nontemporal_load(x + rowBase + N + cc1);
                b1.x = in1 ? bx1 : 0.0f;
                b1.y = in1 ? by1 : 0.0f;
            }

            // ---- Two WMMAs (EXEC is full here) ----
            v8f z0 = {}, z1 = {};
            v8f d0 = __builtin_amdgcn_wmma_f32_16x16x4_f32(
                false, a, false, b0, (short)0, z0, false, false);
            v8f d1 = __builtin_amdgcn_wmma_f32_16x16x4_f32(
                false, a, false, b1, (short)0, z1, false, false);

            // ---- Combine pair for full-wave 128B row stores ----
            // Hoist the convergent cross-lane swaps out of the selects so they
            // execute with full EXEC and the selects lower to v_cndmask.
            const float s0 = swap_halves(d1[0]);
            const float s1 = swap_halves(d1[1]);
            const float s2 = swap_halves(d1[2]);
            const float s3 = swap_halves(d1[3]);
            const float r0 = lo ? d0[0] : s0;
            const float r1 = lo ? d0[1] : s1;
            const float r2 = lo ? d0[2] : s2;
            const float r3 = lo ? d0[3] : s3;

            const int colw = (t0 << 4) + lane;   // 32 contiguous columns
            if (colw < N) {
                __builtin_nontemporal_store(r0, out + 0 * N + colw);
                __builtin_nontemporal_store(r1, out + 1 * N + colw);
                __builtin_nontemporal_store(r2, out + 2 * N + colw);
                __builtin_nontemporal_store(r3, out + 3 * N + colw);
            }
        }
    }
}

extern "C" void kernel_launch(void* const* d_in, const int* in_sizes, int n_in,
                              void* d_out, int out_size, void* d_ws, size_t ws_size,
                              hipStream_t stream) {
    const float* x     = (const float*)d_in[0];  // [4, N]
    const float* w     = (const float*)d_in[1];  // [16, 3]
    const float* v     = (const float*)d_in[2];  // [16, 3]
    const float* theta = (const float*)d_in[3];  // [16, 1]
    const int*   ci    = (const int*)d_in[4];    // scalar
    float* out = (float*)d_out;

    const int N = in_sizes[0] / 4;
    const int nTiles = (N + 15) / 16;
    const int threads = 256;                     // 8 waves/block (wave32)
    const int wavesPerBlock = threads / 32;
    const int tilesPerWaveIt = 2 * SE3_PAIRS;
    long long wavesNeeded = ((long long)nTiles + tilesPerWaveIt - 1) / tilesPerWaveIt;
    int blocks = (int)((wavesNeeded + wavesPerBlock - 1) / wavesPerBlock);
    if (blocks > 2048) blocks = 2048;            // grid-stride covers the rest
    if (blocks < 1) blocks = 1;

    se3_wmma_kernel<<<blocks, threads, 0, stream>>>(x, w, v, theta, ci, out, N);
}